// LDSLoss_5463198400818
// MI455X (gfx1250) — compile-verified
//
#include <hip/hip_runtime.h>
#include <hip/hip_bf16.h>
#include <math.h>

#if defined(__HIP_DEVICE_COMPILE__) && defined(__gfx1250__) && \
    __has_builtin(__builtin_amdgcn_global_load_async_to_lds_b32)
#define LDS_ASYNC 1
#else
#define LDS_ASYNC 0
#endif

__global__ void lds_loss_init(double* acc, unsigned* mx) {
    if (threadIdx.x == 0 && blockIdx.x == 0) {
        acc[0] = 0.0;   // S_fin
        acc[1] = 0.0;   // S_inf
        mx[0]  = 0u;    // max finite weight (positive-float bits)
    }
}

__device__ __forceinline__ void proc_sample(
    float x0, float x1, float t0, float t1,
    float e00, float iw0, float e01, float iw1, int nb,
    const float* __restrict__ sm,
    float& sfin, float& sinf, float& wmax)
{
    float d0 = x0 - t0, d1 = x1 - t1;
    float sq = d0 * d0 + d1 * d1;
    int i0 = (int)floorf((t0 - e00) * iw0);
    int i1 = (int)floorf((t1 - e01) * iw1);
    i0 = i0 < 0 ? 0 : (i0 > nb - 1 ? nb - 1 : i0);
    i1 = i1 < 0 ? 0 : (i1 > nb - 1 ? nb - 1 : i1);
    float kw = sm[i0 * nb + i1];
    float w  = 1.0f / kw;
    if (__builtin_isinf(w)) {
        sinf += sq;                 // weight deferred: multiplied by finite_max later
    } else {
        sfin += w * sq;
        wmax  = fmaxf(wmax, w);
    }
}

__global__ void __launch_bounds__(256)
lds_loss_main(const float* __restrict__ input,
              const float* __restrict__ target,
              const float* __restrict__ smoothed,
              const float* __restrict__ bins0,
              const float* __restrict__ bins1,
              int nb, int npair, long long last_odd, // -1 if N even, else sample idx
              double* acc, unsigned* mx)
{
    extern __shared__ float smem[];                 // nb*nb gather table (~39 KB)
    const int tid = threadIdx.x;
    const int bs  = blockDim.x;
    const int nb2 = nb * nb;

    // ---- Stage the smoothed-distribution table into LDS (CDNA5 async path) ----
#if LDS_ASYNC
    for (int j = tid; j < nb2; j += bs) {
        __builtin_amdgcn_global_load_async_to_lds_b32(
            (__attribute__((address_space(1))) int*)(void*)(smoothed + j),
            (__attribute__((address_space(3))) int*)(void*)(smem + j),
            0, 0);
    }
#if __has_builtin(__builtin_amdgcn_s_wait_asynccnt)
    __builtin_amdgcn_s_wait_asynccnt(0);
#else
    asm volatile("s_wait_asynccnt 0" ::: "memory");
#endif
#else
    for (int j = tid; j < nb2; j += bs) smem[j] = smoothed[j];
#endif
    __syncthreads();

    // Uniform bin edges: searchsorted collapses to an affine map + clamp.
    const float e00 = bins0[0];
    const float iw0 = (float)nb / (bins0[nb] - bins0[0]);
    const float e01 = bins1[0];
    const float iw1 = (float)nb / (bins1[nb] - bins1[0]);

    float sfin = 0.0f, sinf = 0.0f, wmax = 0.0f;

    const float4* __restrict__ in4 = (const float4*)input;
    const float4* __restrict__ tg4 = (const float4*)target;
    const int stride = gridDim.x * bs;

    for (int i = blockIdx.x * bs + tid; i < npair; i += stride) {
        __builtin_prefetch(in4 + i + stride, 0, 0);   // global_prefetch_b8
        __builtin_prefetch(tg4 + i + stride, 0, 0);
        float4 a = in4[i];    // samples 2i and 2i+1 of input  (b128 loads)
        float4 b = tg4[i];    // samples 2i and 2i+1 of target
        proc_sample(a.x, a.y, b.x, b.y, e00, iw0, e01, iw1, nb, smem, sfin, sinf, wmax);
        proc_sample(a.z, a.w, b.z, b.w, e00, iw0, e01, iw1, nb, smem, sfin, sinf, wmax);
    }
    if (last_odd >= 0 && blockIdx.x == 0 && tid == 0) {
        long long s = last_odd;
        proc_sample(input[2*s], input[2*s+1], target[2*s], target[2*s+1],
                    e00, iw0, e01, iw1, nb, smem, sfin, sinf, wmax);
    }

    // ---- wave32 reduction ----
    for (int off = 16; off > 0; off >>= 1) {
        sfin += __shfl_down(sfin, off, 32);
        sinf += __shfl_down(sinf, off, 32);
        wmax  = fmaxf(wmax, __shfl_down(wmax, off, 32));
    }

    // ---- cross-wave reduction (8 waves per 256-thread block) ----
    __shared__ float r_fin[8], r_inf[8], r_max[8];
    const int wid = tid >> 5, lane = tid & 31;
    if (lane == 0) { r_fin[wid] = sfin; r_inf[wid] = sinf; r_max[wid] = wmax; }
    __syncthreads();
    if (tid == 0) {
        double F = 0.0, I = 0.0;
        float  M = 0.0f;
        const int nw = bs >> 5;
        for (int w = 0; w < nw; ++w) {
            F += (double)r_fin[w];
            I += (double)r_inf[w];
            M  = fmaxf(M, r_max[w]);
        }
        atomicAdd(&acc[0], F);
        atomicAdd(&acc[1], I);
        atomicMax(mx, __float_as_uint(M));   // weights > 0: uint order == float order
    }
}

__global__ void lds_loss_finalize(const double* acc, const unsigned* mx,
                                  const float* scaling, float* out, double inv_count)
{
    if (threadIdx.x == 0 && blockIdx.x == 0) {
        float  M    = __uint_as_float(mx[0]);
        double loss = (double)scaling[0] * (acc[0] + (double)M * acc[1]) * inv_count;
        out[0] = (float)loss;
    }
}

extern "C" void kernel_launch(void* const* d_in, const int* in_sizes, int n_in,
                              void* d_out, int out_size, void* d_ws, size_t ws_size,
                              hipStream_t stream) {
    const float* input    = (const float*)d_in[0];
    const float* target   = (const float*)d_in[1];
    const float* smoothed = (const float*)d_in[2];
    const float* bins0    = (const float*)d_in[3];
    const float* bins1    = (const float*)d_in[4];
    const float* scaling  = (const float*)d_in[5];

    const long long n2 = (long long)in_sizes[0];   // 2N flat elements
    const long long N  = n2 / 2;
    int nb = in_sizes[3] - 1;                      // edges -> bins
    if (nb < 1) nb = 1;

    double*   acc = (double*)d_ws;
    unsigned* mx  = (unsigned*)((char*)d_ws + 16);

    const int npair = (int)(N >> 1);
    const long long last_odd = (N & 1) ? (N - 1) : -1;

    int threads = 256;
    int blocks  = (npair + threads - 1) / threads;
    if (blocks > 4096) blocks = 4096;
    if (blocks < 1)    blocks = 1;
    size_t shmem = (size_t)nb * (size_t)nb * sizeof(float);

    lds_loss_init<<<1, 1, 0, stream>>>(acc, mx);
    lds_loss_main<<<blocks, threads, shmem, stream>>>(
        input, target, smoothed, bins0, bins1, nb, npair, last_odd, acc, mx);
    lds_loss_finalize<<<1, 1, 0, stream>>>(acc, mx, scaling, (float*)d_out,
                                           1.0 / (double)n2);
}